// ClassWiseResponseMemory_48120813585096
// MI455X (gfx1250) — compile-verified
//
#include <hip/hip_runtime.h>

// ---------------------------------------------------------------------------
// ClassWiseResponseMemory on MI455X (gfx1250, wave32).
//
// Reference semantics (MOMENTUM=0.1, START=0, UPDATE_INTERVAL=1):
//   per sample i (batch order): c = counts[t_i]; init if c<=0;
//   new = init ? r_i : 0.9*old + 0.1*r_i; memory[t_i] = new; means[i] = new.
// Chains are per-class sequential; classes independent.
//
// We regroup by class and evaluate each chain chunk (<=16 occurrences) as a
// lower-triangular 16x16 weight matrix times a 16x16 response tile using
// chained V_WMMA_F32_16X16X4_F32 (fp32 throughout), plus a rank-1 "carry"
// term for the previous chunk's last mean / the memory bank row.
// Memory-bound: ~72 MB total traffic -> ~3.1 us at 23.3 TB/s.
// ---------------------------------------------------------------------------

typedef __attribute__((ext_vector_type(2))) float v2f;
typedef __attribute__((ext_vector_type(8))) float v8f;

#define CW_MOM 0.1f
#define CW_START 0
#define CW_MAX_OCC 4096   // >= B in the harness

__device__ __forceinline__ float cw_pow09(int e) {
  float p = 1.0f;
#pragma unroll 1
  for (int i = 0; i < e; ++i) p *= (1.0f - CW_MOM);
  return p;
}

// W[k][j] for the chunk's triangular weight matrix.
__device__ __forceinline__ float cw_wcoef(int k, int j, bool initc) {
  if (j > k) return 0.0f;
  if (initc && j == 0) return cw_pow09(k);     // init chain: r_0 passes with 0.9^k
  return CW_MOM * cw_pow09(k - j);             // 0.1 * 0.9^(k-j)
}

// ---------------------------------------------------------------------------
// Prep kernels
// ---------------------------------------------------------------------------

// rank[i] = #{ j < i : targets[j] == targets[i] }  (stable within-class order)
__global__ void cwrm_rank(const int* __restrict__ targets, int* __restrict__ rank, int B) {
  int i = blockIdx.x * blockDim.x + threadIdx.x;
  if (i >= B) return;
  int ti = targets[i];
  int r = 0;
  for (int j = 0; j < i; ++j) r += (targets[j] == ti) ? 1 : 0;
  rank[i] = r;
}

// hist[c] = occurrences of class c in the batch
__global__ void cwrm_hist(const int* __restrict__ targets, int* __restrict__ hist, int B, int C) {
  int c = blockIdx.x * blockDim.x + threadIdx.x;
  if (c >= C) return;
  int cnt = 0;
  for (int j = 0; j < B; ++j) cnt += (targets[j] == c) ? 1 : 0;
  hist[c] = cnt;
}

// exclusive prefix sum over hist -> offs (single block, chunked Hillis-Steele)
__global__ void cwrm_scan(const int* __restrict__ hist, int* __restrict__ offs, int C) {
  __shared__ int s[1024];
  __shared__ int carry;
  if (threadIdx.x == 0) carry = 0;
  __syncthreads();
  for (int base = 0; base < C; base += 1024) {
    int i = base + (int)threadIdx.x;
    int v = (i < C) ? hist[i] : 0;
    s[threadIdx.x] = v;
    __syncthreads();
    for (int off = 1; off < 1024; off <<= 1) {
      int t = (threadIdx.x >= (unsigned)off) ? s[threadIdx.x - off] : 0;
      __syncthreads();
      s[threadIdx.x] += t;
      __syncthreads();
    }
    int incl = s[threadIdx.x];
    if (i < C) offs[i] = incl - v + carry;
    __syncthreads();
    if (threadIdx.x == 1023) carry += s[1023];
    __syncthreads();
  }
}

// idxl[offs[t_i] + rank_i] = i  (per-class, batch-ordered sample lists)
__global__ void cwrm_scatter(const int* __restrict__ targets, const int* __restrict__ rank,
                             const int* __restrict__ offs, int* __restrict__ idxl, int B) {
  int i = blockIdx.x * blockDim.x + threadIdx.x;
  if (i >= B) return;
  idxl[offs[targets[i]] + rank[i]] = i;
}

// ---------------------------------------------------------------------------
// Specialized feature-tile loop: STEPS known at compile time so the hot loop
// is branch-free: one load clause (2*STEPS B-gathers + old), one WMMA chain,
// carry FMAs, predicated stores.
// ---------------------------------------------------------------------------
template <int STEPS>
__device__ __forceinline__ void cw_tiles(
    const float* __restrict__ responses,
    const float* __restrict__ oldBase,
    float* __restrict__ means,
    const size_t* rbA, const size_t* rbB, const size_t* sb,
    const v2f* A, const float* cOld, const bool* valid,
    int wave, int waves, int nTiles, int n) {
  for (int t = wave; t < nTiles; t += waves) {
    const int f = t * 16 + n;

    // Batch all loads so they issue as one clause with a single wait.
    const float oldv = oldBase[f];
    v2f Bv[STEPS];
#pragma unroll
    for (int s = 0; s < STEPS; ++s) {
      Bv[s].x = responses[rbA[s] + f];
      Bv[s].y = responses[rbB[s] + f];
    }

    v8f d = {};
#pragma unroll
    for (int s = 0; s < STEPS; ++s)
      d = __builtin_amdgcn_wmma_f32_16x16x4_f32(false, A[s], false, Bv[s],
                                                (short)0, d, false, false);

    // carry term: d[row] += 0.9^(row+1) * old   (zero coeff when initializing)
#pragma unroll
    for (int v = 0; v < 8; ++v) d[v] += cOld[v] * oldv;

    // D layout: VGPR v -> row (v + 8*hi), col n. Store only valid rows.
#pragma unroll
    for (int v = 0; v < 8; ++v)
      if (valid[v]) means[sb[v] + f] = d[v];
  }
}

// ---------------------------------------------------------------------------
// Main kernel: one block (8 waves) per class; WMMA chain evaluation.
// ---------------------------------------------------------------------------
__global__ __launch_bounds__(256) void cwrm_ema_wmma(
    const float* __restrict__ responses,
    const float* __restrict__ memory,
    const int* __restrict__ counts,
    const int* __restrict__ hist,
    const int* __restrict__ offs,
    const int* __restrict__ idxl,
    float* __restrict__ means,
    int Fdim) {
  const int c = blockIdx.x;
  const int m = hist[c];
  if (m == 0) return;

  __shared__ int sIdx[CW_MAX_OCC];
  const int base = offs[c];
  for (int k = threadIdx.x; k < m; k += blockDim.x) sIdx[k] = idxl[base + k];
  __syncthreads();

  const int cnt0 = counts[c];
  const bool init0 = (cnt0 <= CW_START);   // first occurrence initializes

  const int lane  = threadIdx.x & 31;
  const int wave  = threadIdx.x >> 5;
  const int waves = (int)blockDim.x >> 5;
  const int n  = lane & 15;                // column within 16-wide tile / row M for A
  const int hi = lane >> 4;                // half-wave selector (K/M split per ISA layout)
  const int nTiles  = Fdim >> 4;
  const int nChunks = (m + 15) >> 4;

  for (int chunk = 0; chunk < nChunks; ++chunk) {
    const int mc = min(m - chunk * 16, 16);     // valid occurrences in this chunk (>=1)
    const int steps = (mc + 3) >> 2;            // # of K=4 WMMA steps actually needed
    const bool initc = init0 && (chunk == 0);
    const int kb = chunk * 16;

    // A (16x4 per step): lane L holds M = L&15; VGPR0 = K(2*hi + 4s), VGPR1 = next K.
    v2f A[4];
#pragma unroll
    for (int s = 0; s < 4; ++s) {
      A[s].x = cw_wcoef(n, 4 * s + 2 * hi,     initc);
      A[s].y = cw_wcoef(n, 4 * s + 2 * hi + 1, initc);
    }

    // carry-term coefficient and store validity per D row (row = v + 8*hi)
    float cOld[8];
    bool  valid[8];
#pragma unroll
    for (int v = 0; v < 8; ++v) {
      const int row = v + 8 * hi;
      cOld[v]  = initc ? 0.0f : cw_pow09(row + 1);
      valid[v] = row < mc;
    }

    // per-chunk row base offsets (clamped rows contribute with zero weight)
    size_t rbA[4], rbB[4];
#pragma unroll
    for (int s = 0; s < 4; ++s) {
      const int rA = 4 * s + 2 * hi;
      rbA[s] = (size_t)sIdx[kb + min(rA,     mc - 1)] * (size_t)Fdim;
      rbB[s] = (size_t)sIdx[kb + min(rA + 1, mc - 1)] * (size_t)Fdim;
    }
    size_t sb[8];
#pragma unroll
    for (int v = 0; v < 8; ++v)
      sb[v] = (size_t)sIdx[kb + min(v + 8 * hi, mc - 1)] * (size_t)Fdim;

    // "old" vector: memory bank row for chunk 0, else last mean of previous chunk
    // (written by this same wave in the previous chunk iteration -> HW-ordered).
    const float* oldBase = (chunk == 0)
        ? (memory + (size_t)c * (size_t)Fdim)
        : (means + (size_t)sIdx[kb - 1] * (size_t)Fdim);

    // steps is block-uniform: pick a branch-free specialized hot loop.
    switch (steps) {
      case 1: cw_tiles<1>(responses, oldBase, means, rbA, rbB, sb, A, cOld, valid,
                          wave, waves, nTiles, n); break;
      case 2: cw_tiles<2>(responses, oldBase, means, rbA, rbB, sb, A, cOld, valid,
                          wave, waves, nTiles, n); break;
      case 3: cw_tiles<3>(responses, oldBase, means, rbA, rbB, sb, A, cOld, valid,
                          wave, waves, nTiles, n); break;
      default: cw_tiles<4>(responses, oldBase, means, rbA, rbB, sb, A, cOld, valid,
                           wave, waves, nTiles, n); break;
    }
  }
}

// ---------------------------------------------------------------------------
// Launch
// ---------------------------------------------------------------------------
extern "C" void kernel_launch(void* const* d_in, const int* in_sizes, int n_in,
                              void* d_out, int out_size, void* d_ws, size_t ws_size,
                              hipStream_t stream) {
  (void)n_in; (void)out_size; (void)ws_size;
  const float* responses = (const float*)d_in[0];
  const int*   targets   = (const int*)d_in[1];
  const float* memory    = (const float*)d_in[2];
  const int*   counts    = (const int*)d_in[3];
  float* means = (float*)d_out;

  const int B = in_sizes[1];             // 4096
  const int C = in_sizes[3];             // 1000
  const int F = in_sizes[0] / B;         // 2048

  int* rank = (int*)d_ws;                // B
  int* hist = rank + B;                  // C
  int* offs = hist + C;                  // C
  int* idxl = offs + C;                  // B

  cwrm_rank<<<(B + 255) / 256, 256, 0, stream>>>(targets, rank, B);
  cwrm_hist<<<(C + 255) / 256, 256, 0, stream>>>(targets, hist, B, C);
  cwrm_scan<<<1, 1024, 0, stream>>>(hist, offs, C);
  cwrm_scatter<<<(B + 255) / 256, 256, 0, stream>>>(targets, rank, offs, idxl, B);
  cwrm_ema_wmma<<<C, 256, 0, stream>>>(responses, memory, counts, hist, offs, idxl, means, F);
}